// BaseVAE_10144712753628
// MI455X (gfx1250) — compile-verified
//
#include <hip/hip_runtime.h>
#include <math.h>

typedef __attribute__((ext_vector_type(2))) float v2f;
typedef __attribute__((ext_vector_type(8))) float v8f;

#define EPS_F 1e-8f
#define TWO_PI_OVER_12 0.5235987755982988f

__device__ __forceinline__ float wave_reduce(float v) {
  #pragma unroll
  for (int off = 16; off > 0; off >>= 1) v += __shfl_down(v, off, 32);
  return v;
}

// ---------------------------------------------------------------- init / kld
__global__ void init_ws_kernel(float* ws) {
  if (threadIdx.x < 3) ws[threadIdx.x] = 0.0f;
}

__global__ void kld_kernel(const float* __restrict__ mean,
                           const float* __restrict__ logvar,
                           float* __restrict__ ws) {
  float acc = 0.0f;
  for (int i = threadIdx.x; i < 32 * 128; i += blockDim.x) {
    float lv = logvar[i];
    float m  = mean[i];
    acc += 1.0f + lv - expf(lv) - m * m;
  }
  acc = wave_reduce(acc);
  __shared__ float part[8];
  int lane = threadIdx.x & 31, wv = threadIdx.x >> 5;
  if (lane == 0) part[wv] = acc;
  __syncthreads();
  if (threadIdx.x == 0) {
    float t = 0.0f;
    #pragma unroll
    for (int i = 0; i < 8; ++i) t += part[i];
    ws[0] = -0.5f * t;   // sum over batch of kld
  }
}

// ---------------------------------------------------------------- SSIM
// Separable 11x11 Gaussian, valid conv -> 150x150 per (b,c).
// One block = 16x16 output tile of one (b,c).
__global__ __launch_bounds__(256) void ssim_kernel(const float* __restrict__ x,
                                                   const float* __restrict__ y,
                                                   float* __restrict__ ws) {
  const int tileX = (blockIdx.x % 10) * 16;
  const int tileY = (blockIdx.x / 10) * 16;
  const int c = blockIdx.y;
  const int b = blockIdx.z;
  const int tid = threadIdx.x;

  __shared__ float xs[26][26];
  __shared__ float ys2[26][26];
  __shared__ float hm[5][26][16];

  // normalized 1-D gaussian weights (sigma=1.5, size=11)
  float wt[11];
  {
    float s = 0.0f;
    #pragma unroll
    for (int i = 0; i < 11; ++i) {
      float t = ((float)i - 5.0f) / 1.5f;
      wt[i] = expf(-0.5f * t * t);
      s += wt[i];
    }
    float inv = 1.0f / s;
    #pragma unroll
    for (int i = 0; i < 11; ++i) wt[i] *= inv;
  }

  for (int idx = tid; idx < 26 * 26; idx += 256) {
    int r = idx / 26, cc = idx - r * 26;
    int gy = tileY + r, gx = tileX + cc;
    float xv = 0.0f, yv = 0.0f;
    if (gy < 160 && gx < 160) {
      size_t base = (((size_t)b * 160 + gy) * 160 + gx) * 3 + c;
      xv = x[base];
      yv = y[base];
    }
    xs[r][cc] = xv;
    ys2[r][cc] = yv;
  }
  __syncthreads();

  // horizontal pass: 26 rows x 16 output cols, 5 maps
  for (int idx = tid; idx < 26 * 16; idx += 256) {
    int r = idx / 16, cc = idx - r * 16;
    float sx = 0, sy = 0, sxx = 0, syy = 0, sxy = 0;
    #pragma unroll
    for (int j = 0; j < 11; ++j) {
      float xv = xs[r][cc + j], yv = ys2[r][cc + j], wj = wt[j];
      sx  += wj * xv;
      sy  += wj * yv;
      sxx += wj * xv * xv;
      syy += wj * yv * yv;
      sxy += wj * xv * yv;
    }
    hm[0][r][cc] = sx;  hm[1][r][cc] = sy;
    hm[2][r][cc] = sxx; hm[3][r][cc] = syy;
    hm[4][r][cc] = sxy;
  }
  __syncthreads();

  // vertical pass + SSIM formula
  float val = 0.0f;
  {
    int ty = tid / 16, tx = tid & 15;
    int oy = tileY + ty, ox = tileX + tx;
    if (oy < 150 && ox < 150) {
      float mx = 0, my = 0, exx = 0, eyy = 0, exy = 0;
      #pragma unroll
      for (int i = 0; i < 11; ++i) {
        float wi = wt[i];
        mx  += wi * hm[0][ty + i][tx];
        my  += wi * hm[1][ty + i][tx];
        exx += wi * hm[2][ty + i][tx];
        eyy += wi * hm[3][ty + i][tx];
        exy += wi * hm[4][ty + i][tx];
      }
      const float c1 = 0.0001f;   // (0.01)^2
      const float c2 = 0.0009f;   // (0.03)^2
      float vx  = exx - mx * mx;
      float vy  = eyy - my * my;
      float cov = exy - mx * my;
      float lum = (2.0f * mx * my + c1) / (mx * mx + my * my + c1);
      float cs  = (2.0f * cov + c2) / (vx + vy + c2);
      val = lum * cs;
    }
  }
  val = wave_reduce(val);
  __shared__ float part[8];
  int lane = tid & 31, wv = tid >> 5;
  if (lane == 0) part[wv] = val;
  __syncthreads();
  if (tid == 0) {
    float t = 0.0f;
    #pragma unroll
    for (int i = 0; i < 8; ++i) t += part[i];
    atomicAdd(&ws[1], t);   // raw sum of ssim map over everything
  }
}

// ---------------------------------------------------------------- STFT
// One block = one (b, c, patch-row py) band: image rows [4*py, 4*py+12) x 160 cols.
// Each wave handles patches px = wave, wave+8, ... via two chained f32 WMMA
// stages (16x16x12 each, K split into 3x K=4).
//   A (constant, both stages) = [Wr; Wi] stacked  (rows 0-5 = cos, 6-11 = -sin)
//   stage 1: D1 = A x P            (T = W*P, stacked re/im)
//   stage 2: D2 = A x D1^T         (F^T combos)
//   re F[u][v] = D2[v][u]   - D2[v+6][u+6]
//   im F[u][v] = D2[v][u+6] + D2[v+6][u]
//
// No masking needed on B fragments:
//  - stage-1 lanes n=12..15 pollute only D1 columns 12..15, which stage 2
//    never reads (its K index <= 11); band arrays padded to 164 cols so the
//    unconditional read stays in-bounds.
//  - stage-2 lanes n=12..15 read D1 rows 12..15 which are exactly zero
//    (A rows 12..15 are zero).
__global__ __launch_bounds__(256) void stft_kernel(const float* __restrict__ xin,
                                                   const float* __restrict__ xout,
                                                   float* __restrict__ ws) {
  const int py = blockIdx.x;   // 0..37
  const int c  = blockIdx.y;   // 0..2
  const int b  = blockIdx.z;   // 0..31
  const int tid  = threadIdx.x;
  const int lane = tid & 31;
  const int w    = tid >> 5;

  __shared__ float pin_s[12][164];    // padded: cols 160..163 zero
  __shared__ float pout_s[12][164];
  __shared__ float stgT[8][16][17];   // per-wave transpose staging
  __shared__ float stgO[8][16][17];   // D2 for x_out
  __shared__ float stgI[8][16][17];   // D2 for x_in

  for (int idx = tid; idx < 12 * 164; idx += 256) {
    int r = idx / 164, cc = idx - r * 164;
    float vi = 0.0f, vo = 0.0f;
    if (cc < 160) {
      size_t base = (((size_t)b * 160 + (py * 4 + r)) * 160 + cc) * 3 + c;
      vi = xin[base];
      vo = xout[base];
    }
    pin_s[r][cc]  = vi;
    pout_s[r][cc] = vo;
  }
  __syncthreads();

  // A-matrix fragments: element (m, k); lane holds m = lane&15,
  // k = (lane&16 ? 2 : 0) + vgpr + 4*s   (per ISA 16x4 f32 A layout)
  const int m   = lane & 15;
  const int khi = (lane & 16) ? 2 : 0;
  v2f Afrag[3];
  #pragma unroll
  for (int s = 0; s < 3; ++s) {
    #pragma unroll
    for (int j = 0; j < 2; ++j) {
      int k = 4 * s + khi + j;
      float ang = TWO_PI_OVER_12 * (float)((m % 6) * k);
      float val = 0.0f;
      if (m < 6)       val =  cosf(ang);   // Wr
      else if (m < 12) val = -sinf(ang);   // Wi
      Afrag[s][j] = val;
    }
  }

  const int n = lane & 15;
  const int mhi = (lane & 16) ? 8 : 0;
  float acc = 0.0f;

  for (int px = w; px < 38; px += 8) {
    const int col0 = px * 4;

    #pragma unroll
    for (int img = 0; img < 2; ++img) {
      const float (*band)[164] = img ? pin_s : pout_s;

      // ---- stage 1: T = A x P  (unconditional LDS reads, see note above)
      v8f d1 = {};
      #pragma unroll
      for (int s = 0; s < 3; ++s) {
        v2f bfr;
        #pragma unroll
        for (int j = 0; j < 2; ++j) {
          int k = 4 * s + khi + j;               // patch row, always < 12
          bfr[j] = band[k][col0 + n];
        }
        d1 = __builtin_amdgcn_wmma_f32_16x16x4_f32(
                 false, Afrag[s], false, bfr, (short)0, d1, false, false);
      }

      // stage T (D-layout) to per-wave LDS
      #pragma unroll
      for (int r = 0; r < 8; ++r) stgT[w][r + mhi][n] = d1[r];
      __builtin_amdgcn_wave_barrier();

      // ---- stage 2: D2 = A x T^T   (B element (k,n) = T[n][k])
      v8f d2 = {};
      #pragma unroll
      for (int s = 0; s < 3; ++s) {
        v2f bfr;
        #pragma unroll
        for (int j = 0; j < 2; ++j) {
          int k = 4 * s + khi + j;               // < 12; pair not always 8B aligned (row stride 17)
          bfr[j] = stgT[w][n][k];
        }
        d2 = __builtin_amdgcn_wmma_f32_16x16x4_f32(
                 false, Afrag[s], false, bfr, (short)0, d2, false, false);
      }
      #pragma unroll
      for (int r = 0; r < 8; ++r) {
        if (img) stgI[w][r + mhi][n] = d2[r];
        else     stgO[w][r + mhi][n] = d2[r];
      }
      __builtin_amdgcn_wave_barrier();
    }

    // extract 6x6 half spectra; weight ff[u][v] = (u/5)*(v/5)
    for (int t = lane; t < 36; t += 32) {
      int u = t / 6, v = t - u * 6;
      float ro = stgO[w][v][u]     - stgO[w][v + 6][u + 6];
      float io = stgO[w][v][u + 6] + stgO[w][v + 6][u];
      float ri = stgI[w][v][u]     - stgI[w][v + 6][u + 6];
      float ii = stgI[w][v][u + 6] + stgI[w][v + 6][u];
      float argd = fabsf(atan2f(io, ro + EPS_F) - atan2f(ii, ri + EPS_F));
      float ampd = fabsf(sqrtf(ro * ro + io * io) - sqrtf(ri * ri + ii * ii));
      float wgt = (float)(u * v) * (1.0f / 25.0f);
      acc += wgt * (argd + ampd);
    }
  }

  acc = wave_reduce(acc);
  if (lane == 0) atomicAdd(&ws[2], acc);
}

// ---------------------------------------------------------------- combine
__global__ void combine_kernel(const float* __restrict__ ws,
                               float* __restrict__ out) {
  if (threadIdx.x == 0) {
    const float inv_pix = 1.0f / (150.0f * 150.0f * 3.0f);
    // BETA=1, SSIM_COEF=1, STFT_COEF=1e-4, batch mean /32
    out[0] = (ws[0] + ws[1] * inv_pix + 1e-4f * ws[2]) * (1.0f / 32.0f);
  }
}

extern "C" void kernel_launch(void* const* d_in, const int* in_sizes, int n_in,
                              void* d_out, int out_size, void* d_ws, size_t ws_size,
                              hipStream_t stream) {
  const float* mean   = (const float*)d_in[0];
  const float* logvar = (const float*)d_in[1];
  const float* x_in   = (const float*)d_in[2];
  const float* x_out  = (const float*)d_in[3];
  float* ws  = (float*)d_ws;
  float* out = (float*)d_out;

  init_ws_kernel<<<1, 32, 0, stream>>>(ws);
  kld_kernel<<<1, 256, 0, stream>>>(mean, logvar, ws);
  ssim_kernel<<<dim3(100, 3, 32), 256, 0, stream>>>(x_in, x_out, ws);
  stft_kernel<<<dim3(38, 3, 32), 256, 0, stream>>>(x_in, x_out, ws);
  combine_kernel<<<1, 32, 0, stream>>>(ws, out);
}